// Net_2448131359245
// MI455X (gfx1250) — compile-verified
//
#include <hip/hip_runtime.h>
#include <math.h>

#define N_NODES   50000
#define N_EDGES   300000
#define D_FEAT    256
#define N_CLASSES 256

typedef __attribute__((ext_vector_type(2))) float v2f;
typedef __attribute__((ext_vector_type(8))) float v8f;

// Monotonic float -> uint encoding so a u32 atomic-max implements float max.
// key(-inf) = 0x007FFFFF; anything real compares greater.
#define KEY_NEG_INF 0x007FFFFFu

__device__ __forceinline__ unsigned fenc(float f) {
  unsigned u = __float_as_uint(f);
  return (u & 0x80000000u) ? ~u : (u | 0x80000000u);
}
__device__ __forceinline__ float fdec(unsigned k) {
  unsigned u = (k & 0x80000000u) ? (k ^ 0x80000000u) : ~k;
  return __uint_as_float(u);
}

// ---------------- Kernel 1: init agg keys to encoding of -inf ----------------
__global__ void init_keys_kernel(unsigned* __restrict__ keys, int n4) {
  int i = blockIdx.x * blockDim.x + threadIdx.x;
  if (i < n4) {
    uint4 v;
    v.x = v.y = v.z = v.w = KEY_NEG_INF;
    ((uint4*)keys)[i] = v;
  }
}

// ---------------- Kernel 2: edge scatter-max (gather x[src], atomic-max into dst) ----
// 64 threads per edge, 4 feats per thread -> coalesced float4 gathers, u32 atomics to L2.
__global__ void scatter_max_kernel(const float* __restrict__ x,
                                   const long long* __restrict__ ei,
                                   unsigned* __restrict__ keys) {
  int t = blockIdx.x * blockDim.x + threadIdx.x;
  int e = t >> 6;
  if (e >= N_EDGES) return;
  int c = (t & 63) << 2;
  int src = (int)ei[e];
  int dst = (int)ei[N_EDGES + e];
  const float4 v = *(const float4*)(x + src * D_FEAT + c);
  unsigned* kp = keys + (size_t)dst * D_FEAT + c;
  atomicMax(kp + 0, fenc(v.x));
  atomicMax(kp + 1, fenc(v.y));
  atomicMax(kp + 2, fenc(v.z));
  atomicMax(kp + 3, fenc(v.w));
}

// ---------------- Kernel 3: fused dual-GEMM (WMMA f32) + bias + log_softmax ----------
// One block = 16 output rows x 256 cols. 8 waves, each owns two 16x16 N-tiles.
#define AS (D_FEAT + 2)   // LDS stride for A staging (258 -> conflict-free column reads)
#define TS 257            // LDS stride for output tile

__global__ __launch_bounds__(256) void sage_gemm_lsm_kernel(
    const unsigned* __restrict__ keys,
    const float* __restrict__ x,
    const float* __restrict__ Wl,
    const float* __restrict__ bl,
    const float* __restrict__ Wr,
    float* __restrict__ out) {
  __shared__ float ldsA[16 * AS];  // decoded agg rows (also reused as output tile)
  __shared__ float ldsX[16 * AS];  // x rows
  __shared__ float lse_s[16];

  const int tid = threadIdx.x;
  const int m0 = blockIdx.x * 16;

  // Stage 16 rows of agg (decoded, -inf -> 0) and x into LDS, fully coalesced.
  for (int j = 0; j < 16; ++j) {
    unsigned k = keys[(size_t)(m0 + j) * D_FEAT + tid];
    ldsA[j * AS + tid] = (k == KEY_NEG_INF) ? 0.0f : fdec(k);
    ldsX[j * AS + tid] = x[(size_t)(m0 + j) * D_FEAT + tid];
  }
  __syncthreads();

  const int wave = tid >> 5;
  const int lane = tid & 31;
  const int l16 = lane & 15;
  const int khalf = (lane >> 4) << 1;  // 0 (lanes 0-15) or 2 (lanes 16-31)
  const int n0 = wave * 32;
  const int n1 = n0 + 16;

  v8f c0 = {};
  v8f c1 = {};

  for (int k = 0; k < D_FEAT; k += 4) {
    const int kr = k + khalf;
    // A fragments (16x4): lane<16 holds (row=l16, K=k..k+1); lane>=16 holds K=k+2..k+3
    v2f aA, aX;
    aA.x = ldsA[l16 * AS + kr];
    aA.y = ldsA[l16 * AS + kr + 1];
    aX.x = ldsX[l16 * AS + kr];
    aX.y = ldsX[l16 * AS + kr + 1];
    // B fragments (4x16): lane<16 holds (K=k..k+1, col=n+l16); lane>=16 K=k+2..k+3
    v2f bL0, bL1, bR0, bR1;
    bL0.x = Wl[kr * N_CLASSES + n0 + l16];
    bL0.y = Wl[(kr + 1) * N_CLASSES + n0 + l16];
    bL1.x = Wl[kr * N_CLASSES + n1 + l16];
    bL1.y = Wl[(kr + 1) * N_CLASSES + n1 + l16];
    bR0.x = Wr[kr * N_CLASSES + n0 + l16];
    bR0.y = Wr[(kr + 1) * N_CLASSES + n0 + l16];
    bR1.x = Wr[kr * N_CLASSES + n1 + l16];
    bR1.y = Wr[(kr + 1) * N_CLASSES + n1 + l16];

    // out = agg@Wl + x@Wr : both accumulate into the same C
    c0 = __builtin_amdgcn_wmma_f32_16x16x4_f32(false, aA, false, bL0, (short)0, c0, false, false);
    c0 = __builtin_amdgcn_wmma_f32_16x16x4_f32(false, aX, false, bR0, (short)0, c0, false, false);
    c1 = __builtin_amdgcn_wmma_f32_16x16x4_f32(false, aA, false, bL1, (short)0, c1, false, false);
    c1 = __builtin_amdgcn_wmma_f32_16x16x4_f32(false, aX, false, bR1, (short)0, c1, false, false);
  }

  const float bias0 = bl[n0 + l16];
  const float bias1 = bl[n1 + l16];

  __syncthreads();  // everyone done reading ldsA/ldsX; safe to reuse as output tile
  float* tile = ldsA;  // 16*TS = 4112 floats <= 16*AS = 4128

  // C/D layout: VGPR v, lanes 0-15 -> (M=v, N=lane); lanes 16-31 -> (M=v+8, N=lane-16)
  const int mbase = (lane >> 4) << 3;
  for (int v = 0; v < 8; ++v) {
    tile[(mbase + v) * TS + n0 + l16] = c0[v] + bias0;
    tile[(mbase + v) * TS + n1 + l16] = c1[v] + bias1;
  }
  __syncthreads();

  // log-softmax: 16 threads per row, shuffle reduce within aligned 16-lane groups
  {
    const int row = tid >> 4;
    const int w16 = tid & 15;
    float vmax = -INFINITY;
    for (int j = 0; j < 16; ++j)
      vmax = fmaxf(vmax, tile[row * TS + w16 + 16 * j]);
    for (int m = 8; m >= 1; m >>= 1)
      vmax = fmaxf(vmax, __shfl_xor(vmax, m, 32));
    float s = 0.0f;
    for (int j = 0; j < 16; ++j)
      s += __expf(tile[row * TS + w16 + 16 * j] - vmax);
    for (int m = 8; m >= 1; m >>= 1)
      s += __shfl_xor(s, m, 32);
    if (w16 == 0) lse_s[row] = vmax + __logf(s);
  }
  __syncthreads();

  // coalesced final store
  for (int i = 0; i < 16; ++i)
    out[(size_t)(m0 + i) * N_CLASSES + tid] = tile[i * TS + tid] - lse_s[i];
}

extern "C" void kernel_launch(void* const* d_in, const int* in_sizes, int n_in,
                              void* d_out, int out_size, void* d_ws, size_t ws_size,
                              hipStream_t stream) {
  const float* x      = (const float*)d_in[0];
  const long long* ei = (const long long*)d_in[1];
  const float* Wl     = (const float*)d_in[2];
  const float* bl     = (const float*)d_in[3];
  const float* Wr     = (const float*)d_in[4];
  float* out = (float*)d_out;
  unsigned* keys = (unsigned*)d_ws;  // N_NODES * D_FEAT u32 = 51.2 MB

  const int n4 = N_NODES * D_FEAT / 4;  // 3.2M uint4 stores
  init_keys_kernel<<<(n4 + 255) / 256, 256, 0, stream>>>(keys, n4);

  scatter_max_kernel<<<(N_EDGES * 64) / 256, 256, 0, stream>>>(x, ei, keys);

  sage_gemm_lsm_kernel<<<N_NODES / 16, 256, 0, stream>>>(keys, x, Wl, bl, Wr, out);
}